// Rf_PredLayer_77077483094325
// MI455X (gfx1250) — compile-verified
//
#include <hip/hip_runtime.h>
#include <math.h>

// ---------------------------------------------------------------------------
// Problem sizes (fixed by the reference)
// ---------------------------------------------------------------------------
#define LAYER_SIZE 65536
#define OUT_SIZE   2048
#define INF_RATE   0.1f

#define ROWS_PER_BLOCK 64      // 4 waves x 16 rows
#define GEMV_THREADS   128     // 4 wave32
#define GEMV_BLOCKS    (LAYER_SIZE / ROWS_PER_BLOCK)   // 1024

typedef __attribute__((ext_vector_type(2))) float v2f;
typedef __attribute__((ext_vector_type(8))) float v8f;

#if defined(__gfx1250__) && __has_builtin(__builtin_amdgcn_wmma_f32_16x16x4_f32)
#define USE_WMMA 1
#else
#define USE_WMMA 0
#endif

// ---------------------------------------------------------------------------
// Kernel 1: e_act = r_out - W @ x ; rpre = r_act + 0.1*(bu - e_act)
// Writes e_act -> out_e, rpre -> out_rpre, per-block (sum, sumsq) partials.
// ---------------------------------------------------------------------------
__global__ __launch_bounds__(GEMV_THREADS)
void k_gemv_epilogue(const float* __restrict__ W,
                     const float* __restrict__ x,
                     const float* __restrict__ r_out,
                     const float* __restrict__ r_act,
                     const float* __restrict__ bu,
                     float* __restrict__ out_e,
                     float* __restrict__ out_rpre,
                     float* __restrict__ psum,
                     float* __restrict__ psumsq)
{
    __shared__ float s_rpre[ROWS_PER_BLOCK];

    const int tid  = threadIdx.x;
    const int wave = tid >> 5;
    const int lane = tid & 31;

#if USE_WMMA
    // ---- WMMA f32 16x16x4 GEMV: B = x broadcast along N ----
    const int hlf   = lane >> 4;          // 0: K slots {0,1}, 1: K slots {2,3}
    const int mrow  = lane & 15;          // row within 16-row tile (A layout)
    const int rbase = blockIdx.x * ROWS_PER_BLOCK + wave * 16;
    const int koff  = hlf * 2;

    const float* wrow = W + (size_t)(rbase + mrow) * OUT_SIZE + koff;
    const float* xv   = x + koff;

    v8f c = {};
    #pragma unroll 4
    for (int k0 = 0; k0 < OUT_SIZE; k0 += 4) {
        v2f a = *(const v2f*)(wrow + k0);   // W[row, k0+koff .. +1]
        v2f b = *(const v2f*)(xv   + k0);   // x[k0+koff .. +1] (bcast over N)
        // D = A*B + C ; every column of D equals dot(W_row, x)
        c = __builtin_amdgcn_wmma_f32_16x16x4_f32(
                /*neg_a=*/false, a, /*neg_b=*/false, b,
                /*c_mod=*/(short)0, c, /*reuse_a=*/false, /*reuse_b=*/false);
    }

    // C/D layout: VGPR j -> row j (lanes 0-15) / row j+8 (lanes 16-31).
    // Columns are identical, so lane 0 has rows 0..7, lane 16 has rows 8..15.
    if (mrow == 0) {
        const int rowb = rbase + hlf * 8;
        #pragma unroll
        for (int j = 0; j < 8; ++j) {
            const int row = rowb + j;
            const float dot = c[j];
            const float e   = r_out[row] - dot;
            const float rp  = r_act[row] + INF_RATE * (bu[row] - e);
            out_e[row]    = e;
            out_rpre[row] = rp;
            s_rpre[wave * 16 + hlf * 8 + j] = rp;
        }
    }
#else
    // ---- Fallback: 2 threads per row, serial FMA halves ----
    __shared__ float s_part[GEMV_THREADS];
    const int row  = blockIdx.x * ROWS_PER_BLOCK + (tid >> 1);
    const int kh   = tid & 1;
    const float* wrow = W + (size_t)row * OUT_SIZE + kh * (OUT_SIZE / 2);
    const float* xv   = x + kh * (OUT_SIZE / 2);
    float acc = 0.0f;
    for (int k = 0; k < OUT_SIZE / 2; ++k) acc = fmaf(wrow[k], xv[k], acc);
    s_part[tid] = acc;
    __syncthreads();
    if (tid < ROWS_PER_BLOCK) {
        const int r = blockIdx.x * ROWS_PER_BLOCK + tid;
        const float dot = s_part[2 * tid] + s_part[2 * tid + 1];
        const float e   = r_out[r] - dot;
        const float rp  = r_act[r] + INF_RATE * (bu[r] - e);
        out_e[r]    = e;
        out_rpre[r] = rp;
        s_rpre[tid] = rp;
    }
#endif

    __syncthreads();
    if (tid == 0) {
        float s = 0.0f, ss = 0.0f;
        #pragma unroll
        for (int i = 0; i < ROWS_PER_BLOCK; ++i) {
            const float v = s_rpre[i];
            s  += v;
            ss += v * v;
        }
        psum[blockIdx.x]   = s;
        psumsq[blockIdx.x] = ss;
    }
}

// ---------------------------------------------------------------------------
// Kernel 2: reduce per-block partials -> threshold = mean + 0.25*std (ddof=1)
// ---------------------------------------------------------------------------
__global__ __launch_bounds__(256)
void k_stats(const float* __restrict__ psum,
             const float* __restrict__ psumsq,
             float* __restrict__ thr_out,
             int nblocks)
{
    __shared__ float ss[256];
    __shared__ float sq[256];
    const int t = threadIdx.x;
    float s = 0.0f, q = 0.0f;
    for (int i = t; i < nblocks; i += 256) {
        s += psum[i];
        q += psumsq[i];
    }
    ss[t] = s; sq[t] = q;
    __syncthreads();
    for (int off = 128; off > 0; off >>= 1) {
        if (t < off) { ss[t] += ss[t + off]; sq[t] += sq[t + off]; }
        __syncthreads();
    }
    if (t == 0) {
        const float N   = (float)LAYER_SIZE;
        const float S   = ss[0];
        const float SS  = sq[0];
        const float mean = S / N;
        float var = (SS - S * S / N) / (N - 1.0f);
        var = var > 0.0f ? var : 0.0f;
        thr_out[0] = mean + 0.25f * sqrtf(var);
    }
}

// ---------------------------------------------------------------------------
// Kernel 3: threshold (in place on r segment) + tanh
// ---------------------------------------------------------------------------
__global__ __launch_bounds__(256)
void k_finish(float* __restrict__ r_seg,
              float* __restrict__ rout_seg,
              const float* __restrict__ thr,
              int n)
{
    const int i = blockIdx.x * blockDim.x + threadIdx.x;
    if (i < n) {
        const float t = thr[0];
        float v = r_seg[i];
        v = (v < t) ? -1.0f : v;
        r_seg[i]    = v;
        rout_seg[i] = tanhf(v);
    }
}

// ---------------------------------------------------------------------------
// Launch
// ---------------------------------------------------------------------------
extern "C" void kernel_launch(void* const* d_in, const int* in_sizes, int n_in,
                              void* d_out, int out_size, void* d_ws, size_t ws_size,
                              hipStream_t stream)
{
    // setup_inputs() order: bu_errors, r_act, r_out, nextlayer_r_out, weights
    const float* bu    = (const float*)d_in[0];
    const float* r_act = (const float*)d_in[1];
    const float* r_out = (const float*)d_in[2];
    const float* x     = (const float*)d_in[3];
    const float* W     = (const float*)d_in[4];

    float* out    = (float*)d_out;
    float* e_seg  = out;                    // e_act
    float* r_seg  = out + LAYER_SIZE;       // r_act (pre, then final in place)
    float* ro_seg = out + 2 * LAYER_SIZE;   // r_out_new

    float* psum   = (float*)d_ws;           // [GEMV_BLOCKS]
    float* psumsq = psum + GEMV_BLOCKS;     // [GEMV_BLOCKS]
    float* thr    = psumsq + GEMV_BLOCKS;   // [1]

    k_gemv_epilogue<<<dim3(GEMV_BLOCKS), dim3(GEMV_THREADS), 0, stream>>>(
        W, x, r_out, r_act, bu, e_seg, r_seg, psum, psumsq);

    k_stats<<<dim3(1), dim3(256), 0, stream>>>(psum, psumsq, thr, GEMV_BLOCKS);

    k_finish<<<dim3((LAYER_SIZE + 255) / 256), dim3(256), 0, stream>>>(
        r_seg, ro_seg, thr, LAYER_SIZE);
}